// ChannelMask_22024592294327
// MI455X (gfx1250) — compile-verified
//
#include <hip/hip_runtime.h>
#include <stdint.h>

// ---------------------------------------------------------------------------
// ChannelMask (point-based-std): per-batch quantile threshold + binary mask.
// Exact 3-level radix select (11/11/10 bits) on order-preserving float keys.
// CDNA5 paths: global_load_async_to_lds_b128 streaming + s_wait_asynccnt,
// LDS (ds_add) privatized histograms, global atomics, NT-hinted float4 vmem.
// L2 strategy: 100MB input fits in the 192MB global L2 and is read 4x; the
// zero-reuse 100MB output is stored with NT hint so it does not evict it.
// ---------------------------------------------------------------------------

#define TPB 256
#define NBINS 2048
#define TILE_ELEMS 1024          // TPB * 4 floats per async tile (4 KB)
#define TILES_PER_BLOCK 8
#define ELEMS_PER_BLOCK 8192     // TILE_ELEMS * TILES_PER_BLOCK

typedef float v4f __attribute__((ext_vector_type(4)));

__device__ __forceinline__ uint32_t f2key(uint32_t u) {
  // monotonic float -> uint transform
  return u ^ ((u & 0x80000000u) ? 0xFFFFFFFFu : 0x80000000u);
}
__device__ __forceinline__ uint32_t key2f(uint32_t k) {
  return k ^ ((k & 0x80000000u) ? 0x80000000u : 0xFFFFFFFFu);
}

// ---------------------------------------------------------------------------
// Init: zero global histograms, set per-batch selection state.
// state[b*8] = {prefix0, rank0, prefix1, rank1, qbits, -, -, -}
// ---------------------------------------------------------------------------
__global__ void cm_init_kernel(uint32_t* __restrict__ state,
                               uint32_t* __restrict__ hist,
                               const int* __restrict__ pr_ptr,
                               int n_per_batch, int bs) {
  int total = bs * 2 * NBINS;
  for (int i = blockIdx.x * blockDim.x + threadIdx.x; i < total;
       i += gridDim.x * blockDim.x)
    hist[i] = 0u;
  if (blockIdx.x == 0 && threadIdx.x < (unsigned)bs) {
    int b = threadIdx.x;
    int pr = *pr_ptr;
    uint64_t k0 = 0;
    if (pr >= 1 && pr <= 9) {
      uint64_t num = (uint64_t)(n_per_batch - 1) * (uint64_t)(10 - pr);
      k0 = num / 10;  // floor((N-1)*(1-0.1*pr))
    }
    uint32_t* st = state + b * 8;
    st[0] = 0u; st[1] = (uint32_t)k0;        // selection for v[k0]
    st[2] = 0u; st[3] = (uint32_t)(k0 + 1);  // selection for v[k0+1]
    st[4] = 0u;
  }
}

// ---------------------------------------------------------------------------
// Histogram pass: stream tiles Global -> LDS with async DMA, double-buffered.
// Two filtered histograms (one per selection state) via LDS atomics.
// Hazard note: the buffer written by async tile t+1 was last ds_load-read at
// iteration t-1, and that data was consumed (s_wait_dscnt) before this asm in
// program order, so WAR on the LDS tile is safe.
// ---------------------------------------------------------------------------
__global__ void cm_hist_kernel(const float* __restrict__ x,
                               const uint32_t* __restrict__ state,
                               uint32_t* __restrict__ hist,
                               int shift, int matchshift, uint32_t binmask,
                               int n_per_batch) {
  __shared__ uint32_t lhist[2 * NBINS];
  __shared__ __align__(16) uint32_t tile[2][TILE_ELEMS];

  const int tid = threadIdx.x;
  const int b = blockIdx.y;

  for (int i = tid; i < 2 * NBINS; i += TPB) lhist[i] = 0u;
  __syncthreads();

  const uint32_t p0 = state[b * 8 + 0];
  const uint32_t p1 = state[b * 8 + 2];

  const float* base = x + (size_t)b * n_per_batch +
                      (size_t)blockIdx.x * ELEMS_PER_BLOCK;

  // issue tile 0 (16 bytes per lane, async global -> LDS)
  {
    uint32_t lds_off = (uint32_t)(uintptr_t)&tile[0][tid * 4];
    uint64_t ga = (uint64_t)(uintptr_t)(base + tid * 4);
    asm volatile("global_load_async_to_lds_b128 %0, %1, off"
                 :: "v"(lds_off), "v"(ga) : "memory");
  }

  for (int t = 0; t < TILES_PER_BLOCK; ++t) {
    if (t + 1 < TILES_PER_BLOCK) {
      uint32_t lds_off = (uint32_t)(uintptr_t)&tile[(t + 1) & 1][tid * 4];
      uint64_t ga = (uint64_t)(uintptr_t)
          (base + (size_t)(t + 1) * TILE_ELEMS + tid * 4);
      asm volatile("global_load_async_to_lds_b128 %0, %1, off"
                   :: "v"(lds_off), "v"(ga) : "memory");
      asm volatile("s_wait_asynccnt 1" ::: "memory");
    } else {
      asm volatile("s_wait_asynccnt 0" ::: "memory");
    }
    // each lane consumes only its own 16-byte slot (no cross-lane barrier)
    const uint32_t* slot = &tile[t & 1][tid * 4];
#pragma unroll
    for (int e = 0; e < 4; ++e) {
      uint32_t key = f2key(slot[e]);
      uint32_t bin = (key >> shift) & binmask;
      bool m0 = (matchshift >= 32) || ((key >> matchshift) == p0);
      bool m1 = (matchshift >= 32) || ((key >> matchshift) == p1);
      if (m0) atomicAdd(&lhist[bin], 1u);            // ds_add_u32
      if (m1) atomicAdd(&lhist[NBINS + bin], 1u);    // ds_add_u32
    }
  }

  __syncthreads();
  uint32_t* gh = hist + (size_t)b * 2 * NBINS;
  for (int i = tid; i < 2 * NBINS; i += TPB) {
    uint32_t v = lhist[i];
    if (v) atomicAdd(&gh[i], v);                     // global_atomic_add_u32
  }
}

// ---------------------------------------------------------------------------
// Scan pass: per batch, find bin containing each rank, refine state,
// zero the global histogram for reuse. At level 3, produce q.
// ---------------------------------------------------------------------------
__global__ void cm_scan_kernel(uint32_t* __restrict__ state,
                               uint32_t* __restrict__ hist,
                               int nbins, int bits, int level,
                               const int* __restrict__ pr_ptr,
                               int n_per_batch) {
  __shared__ uint32_t lh[2 * NBINS];
  const int tid = threadIdx.x;
  const int b = blockIdx.x;
  uint32_t* gh = hist + (size_t)b * 2 * NBINS;
  for (int i = tid; i < 2 * NBINS; i += TPB) { lh[i] = gh[i]; gh[i] = 0u; }
  __syncthreads();

  if (tid == 0) {
    uint32_t* st = state + b * 8;
    for (int sel = 0; sel < 2; ++sel) {
      uint32_t prefix = st[sel * 2 + 0];
      uint32_t rank   = st[sel * 2 + 1];
      uint32_t cum = 0, bin = 0;
      for (int i = 0; i < nbins; ++i) {
        uint32_t c = lh[sel * NBINS + i];
        if (rank < cum + c) { bin = (uint32_t)i; rank -= cum; break; }
        cum += c;
      }
      st[sel * 2 + 0] = (prefix << bits) | bin;
      st[sel * 2 + 1] = rank;
    }
    if (level == 3) {
      int pr = *pr_ptr;
      uint32_t frac_num = 0;
      if (pr >= 1 && pr <= 9) {
        uint64_t num = (uint64_t)(n_per_batch - 1) * (uint64_t)(10 - pr);
        frac_num = (uint32_t)(num % 10);
      }
      float v0 = __uint_as_float(key2f(st[0]));
      float v1 = __uint_as_float(key2f(st[2]));
      float frac = (float)frac_num * 0.1f;
      float q = v0 + frac * (v1 - v0);   // linear interpolation
      st[4] = __float_as_uint(q);
    }
  }
}

// ---------------------------------------------------------------------------
// Mask pass: out = (x >= q[batch]) ? 1 : 0, float4 vectorized.
// Last read of x (NT load) and zero-reuse output (NT store): keeps the input
// resident in the 192MB L2 across the earlier passes.
// pr<=0 -> zeros, pr>=10 -> ones (decided on-device).
// ---------------------------------------------------------------------------
__global__ void cm_mask_kernel(const float* __restrict__ x,
                               float* __restrict__ out,
                               const uint32_t* __restrict__ state,
                               const int* __restrict__ pr_ptr,
                               uint32_t n4_per_batch, uint32_t total4) {
  uint32_t i4 = blockIdx.x * blockDim.x + threadIdx.x;
  if (i4 >= total4) return;
  int pr = *pr_ptr;
  v4f v = __builtin_nontemporal_load((const v4f*)x + i4);
  v4f r;
  if (pr >= 10) {
    r = (v4f){1.0f, 1.0f, 1.0f, 1.0f};
  } else if (pr <= 0) {
    r = (v4f){0.0f, 0.0f, 0.0f, 0.0f};
  } else {
    uint32_t b = i4 / n4_per_batch;
    float q = __uint_as_float(state[b * 8 + 4]);
    r.x = (v.x >= q) ? 1.0f : 0.0f;
    r.y = (v.y >= q) ? 1.0f : 0.0f;
    r.z = (v.z >= q) ? 1.0f : 0.0f;
    r.w = (v.w >= q) ? 1.0f : 0.0f;
  }
  __builtin_nontemporal_store(r, (v4f*)out + i4);
}

// ---------------------------------------------------------------------------
extern "C" void kernel_launch(void* const* d_in, const int* in_sizes, int n_in,
                              void* d_out, int out_size, void* d_ws, size_t ws_size,
                              hipStream_t stream) {
  const float* x = (const float*)d_in[0];
  const int* pr = (const int*)d_in[1];
  float* out = (float*)d_out;

  const int BS = 32;
  const int total = in_sizes[0];      // 32*192*64*64 = 25,165,824
  const int N = total / BS;           // 786,432 per batch

  // workspace: [0,1KB) state, [4KB, 4KB+512KB) histograms
  uint32_t* state = (uint32_t*)d_ws;
  uint32_t* hist  = (uint32_t*)((char*)d_ws + 4096);

  cm_init_kernel<<<64, TPB, 0, stream>>>(state, hist, pr, N, BS);

  const int bpb = N / ELEMS_PER_BLOCK;   // 96 blocks per batch
  dim3 hgrid(bpb, BS);

  // level 1: bits [31:21]
  cm_hist_kernel<<<hgrid, TPB, 0, stream>>>(x, state, hist, 21, 32, 2047u, N);
  cm_scan_kernel<<<BS, TPB, 0, stream>>>(state, hist, 2048, 11, 1, pr, N);
  // level 2: bits [20:10]
  cm_hist_kernel<<<hgrid, TPB, 0, stream>>>(x, state, hist, 10, 21, 2047u, N);
  cm_scan_kernel<<<BS, TPB, 0, stream>>>(state, hist, 2048, 11, 2, pr, N);
  // level 3: bits [9:0]
  cm_hist_kernel<<<hgrid, TPB, 0, stream>>>(x, state, hist, 0, 10, 1023u, N);
  cm_scan_kernel<<<BS, TPB, 0, stream>>>(state, hist, 1024, 10, 3, pr, N);

  const uint32_t total4 = (uint32_t)(total / 4);
  const uint32_t n4 = (uint32_t)(N / 4);
  const int mblocks = (int)((total4 + TPB - 1) / TPB);
  cm_mask_kernel<<<mblocks, TPB, 0, stream>>>(x, out, state, pr, n4, total4);
}